// SubjectBlock_21397527069117
// MI455X (gfx1250) — compile-verified
//
#include <hip/hip_runtime.h>
#include <hip/hip_bf16.h>

// Problem constants (match reference)
#define D1 270
#define TT 2048
#define NB 64
#define NS 32
#define MP 288   // M padded to 18 * 16
#define KP 272   // K padded to 68 * 4

typedef float v2f __attribute__((ext_vector_type(2)));
typedef float v8f __attribute__((ext_vector_type(8)));

// ---------------------------------------------------------------------------
// Kernel 1: fold the shared conv into each subject's conv.
//   Wc[s][m][k] = sum_o subj_w[s][m][o] * conv_w[o][k]   (zero in padded m/k)
// Layout: Wc is [NS][MP][KP] row-major in workspace (~10 MB, L2-resident).
// ---------------------------------------------------------------------------
__global__ __launch_bounds__(256) void build_wc(
    const float* __restrict__ subj_w,   // [NS][D1][D1]
    const float* __restrict__ conv_w,   // [D1][D1]  (o, k)
    float* __restrict__ Wc)             // [NS][MP][KP]
{
    long idx = (long)blockIdx.x * blockDim.x + threadIdx.x;
    const long total = (long)NS * MP * KP;
    if (idx >= total) return;
    int k = (int)(idx % KP);
    int m = (int)((idx / KP) % MP);
    int s = (int)(idx / ((long)KP * MP));
    float acc = 0.0f;
    if (m < D1 && k < D1) {
        const float* srow = subj_w + ((long)s * D1 + m) * D1;
        #pragma unroll 2
        for (int o = 0; o < D1; ++o)
            acc = fmaf(srow[o], conv_w[(long)o * D1 + k], acc);  // coalesced over k
    }
    Wc[idx] = acc;
}

// ---------------------------------------------------------------------------
// Kernel 1b: folded bias  bc[s][m] = sum_o subj_w[s][m][o] * conv_b[o]
// ---------------------------------------------------------------------------
__global__ __launch_bounds__(MP) void build_bc(
    const float* __restrict__ subj_w,
    const float* __restrict__ conv_b,
    float* __restrict__ bc)             // [NS][MP]
{
    int s = blockIdx.x;
    int m = threadIdx.x;
    float acc = 0.0f;
    if (m < D1) {
        const float* srow = subj_w + ((long)s * D1 + m) * D1;
        for (int o = 0; o < D1; ++o)
            acc = fmaf(srow[o], conv_b[o], acc);
    }
    bc[s * MP + m] = acc;
}

// ---------------------------------------------------------------------------
// Kernel 2: out[b] = Wc[idx[b]] @ X[b] + bc[idx[b]]
// Grid: (TT/64, NB). Block: 288 threads = 9 wave32.
// Wave w owns rows [32w, 32w+32) x cols [64*bx, 64*bx+64):
//   2 m-tiles x 4 n-tiles of 16x16 f32 WMMA (8 wmma / K-step).
// Single loop over the padded K range (68 iterations, even trip count so
// the unroller needs no peel -> no accumulator register rotation).  B row
// indices are clamped branch-free; A's zero-padded columns nullify them.
// ---------------------------------------------------------------------------
__global__ __launch_bounds__(MP) void subject_gemm(
    const float* __restrict__ X,      // [NB][D1][TT]
    const int*   __restrict__ sidx,   // [NB]
    const float* __restrict__ Wc,     // [NS][MP][KP]
    const float* __restrict__ bc,     // [NS][MP]
    float* __restrict__ out)          // [NB][D1][TT]
{
    const int b     = blockIdx.y;
    const int s     = sidx[b];
    const int lane  = threadIdx.x & 31;
    const int wave  = threadIdx.x >> 5;     // 0..8
    const int half  = lane >> 4;            // 0 or 1
    const int col   = lane & 15;
    const int mbase = wave * 32;            // this wave's first output row
    const int nbase = blockIdx.x * 64;      // this block's first output col

    const float* A   = Wc + (long)s * MP * KP;
    const float* bcs = bc + s * MP;
    const float* Xb  = X  + (long)b * D1 * TT;
    float*       Ob  = out + (long)b * D1 * TT;

    // Accumulators, initialized with the folded bias (added exactly once).
    // C/D layout: VGPR j, lane l -> row = base + j + 8*(l>>4), col = l&15.
    v8f acc[2][4];
    #pragma unroll
    for (int mi = 0; mi < 2; ++mi) {
        #pragma unroll
        for (int j = 0; j < 8; ++j) {
            float bias = bcs[mbase + 16 * mi + j + 8 * half];
            #pragma unroll
            for (int ni = 0; ni < 4; ++ni)
                acc[mi][ni][j] = bias;
        }
    }

    // A-fragment row for this lane (16x4 f32 A layout: row = lane&15,
    // k = k0 + 2*(lane>>4) .. +1, i.e. one b64 load per m-tile).
    const int arow = mbase + col;
    const float* Arow0 = A + (long)arow * KP + 2 * half;        // m-tile 0
    const float* Arow1 = A + (long)(arow + 16) * KP + 2 * half; // m-tile 1
    const float* Bcol  = Xb + nbase + col;                      // n-tile base

    for (int k0 = 0; k0 < KP; k0 += 4) {
        const int kb = k0 + 2 * half;
        v2f a0 = *(const v2f*)(Arow0 + k0);
        v2f a1 = *(const v2f*)(Arow1 + k0);
        // Clamp padded K rows (268..271 lane-half dependent) so loads never
        // fault; A's zero-padded columns make the products zero anyway.
        const long kr0 = (kb     < D1) ? kb     : (D1 - 1);
        const long kr1 = (kb + 1 < D1) ? kb + 1 : (D1 - 1);
        v2f bfr[4];
        #pragma unroll
        for (int ni = 0; ni < 4; ++ni) {
            bfr[ni].x = Bcol[kr0 * TT + 16 * ni];
            bfr[ni].y = Bcol[kr1 * TT + 16 * ni];
        }
        #pragma unroll
        for (int ni = 0; ni < 4; ++ni) {
            acc[0][ni] = __builtin_amdgcn_wmma_f32_16x16x4_f32(
                false, a0, false, bfr[ni], (short)0, acc[0][ni], false, false);
            acc[1][ni] = __builtin_amdgcn_wmma_f32_16x16x4_f32(
                false, a1, false, bfr[ni], (short)0, acc[1][ni], false, false);
        }
    }

    // Store valid rows only (M padding 270..287 dropped).
    #pragma unroll
    for (int mi = 0; mi < 2; ++mi) {
        #pragma unroll
        for (int j = 0; j < 8; ++j) {
            int row = mbase + 16 * mi + j + 8 * half;
            if (row < D1) {
                float* orow = Ob + (long)row * TT + nbase + col;
                #pragma unroll
                for (int ni = 0; ni < 4; ++ni)
                    orow[16 * ni] = acc[mi][ni][j];
            }
        }
    }
}

// ---------------------------------------------------------------------------
extern "C" void kernel_launch(void* const* d_in, const int* in_sizes, int n_in,
                              void* d_out, int out_size, void* d_ws, size_t ws_size,
                              hipStream_t stream) {
    const float* X      = (const float*)d_in[0];   // [64][270][2048]
    const int*   sidx   = (const int*)  d_in[1];   // [64]
    const float* conv_w = (const float*)d_in[2];   // [270][270]
    const float* conv_b = (const float*)d_in[3];   // [270]
    const float* subj_w = (const float*)d_in[4];   // [32][270][270]
    float*       out    = (float*)d_out;

    float* Wc = (float*)d_ws;                       // NS*MP*KP floats (~10 MB)
    float* bc = Wc + (size_t)NS * MP * KP;          // NS*MP floats

    {
        const long total   = (long)NS * MP * KP;
        const int  threads = 256;
        const unsigned blocks = (unsigned)((total + threads - 1) / threads);
        build_wc<<<dim3(blocks), dim3(threads), 0, stream>>>(subj_w, conv_w, Wc);
    }
    build_bc<<<dim3(NS), dim3(MP), 0, stream>>>(subj_w, conv_b, bc);
    subject_gemm<<<dim3(TT / 64, NB), dim3(MP), 0, stream>>>(X, sidx, Wc, bc, out);
}